// Upsample_23407571763555
// MI455X (gfx1250) — compile-verified
//
#include <hip/hip_runtime.h>
#include <stdint.h>

// upfirdn2d, up=2, pad=(2,1), k = outer([1,3,3,1])^2 normalized * 4, down=1.
// Separable per-dim weights: [0.25, 0.75, 0.75, 0.25]; with zero-insertion each
// output phase is a 2-tap blend:
//   out[2m]   = 0.25*v(m-1) + 0.75*v(m)
//   out[2m+1] = 0.75*v(m)   + 0.25*v(m+1)
// Boundary map on sample grid: i==-1 -> 1, i==H -> H-1.

#define H_IN 256
#define W_IN 256
#define H_OUT 512
#define W_OUT 512
#define TILE_OH 32
#define TILE_OW 128
#define LDS_H 18                 // TILE_OH/2 + 2 (halo)
#define LDS_W 66                 // TILE_OW/2 + 2 (halo)
#define LDS_STRIDE 68
#define LDS_TOT (LDS_H * LDS_W)  // 1188

#if defined(__HIP_DEVICE_COMPILE__) && __has_builtin(__builtin_amdgcn_global_load_async_to_lds_b32)
#define USE_ASYNC_LDS 1
#else
#define USE_ASYNC_LDS 0
#endif

typedef __attribute__((address_space(1))) int as1_int;
typedef __attribute__((address_space(3))) int as3_int;

__device__ __forceinline__ int reflect_idx(int i, int n) {
    // only ever called with i in [-1, n]; -1 -> 1 (reflect past the inserted
    // zero), n -> n-1 (upsampled-grid reflect lands on the last sample)
    return (i < 0) ? 1 : ((i >= n) ? (n - 1) : i);
}

__global__ __launch_bounds__(256) void upfirdn2d_up2_kernel(
    const float* __restrict__ in, float* __restrict__ out) {
    __shared__ float tile[LDS_H * LDS_STRIDE];

    const int tid = threadIdx.x;
    const int img = blockIdx.z;
    const int oy0 = blockIdx.y * TILE_OH;
    const int ox0 = blockIdx.x * TILE_OW;
    const int iy0 = (oy0 >> 1) - 1;  // first (halo) input row staged
    const int ix0 = (ox0 >> 1) - 1;  // first (halo) input col staged

    const float* __restrict__ src = in + (size_t)img * (H_IN * W_IN);

    // ---- Stage 18x66 input tile (with reflect halo) into LDS ------------
    #pragma unroll
    for (int k = 0; k < 5; ++k) {
        int idx = tid + k * 256;
        if (idx < LDS_TOT) {
            int lr = idx / LDS_W;
            int lc = idx - lr * LDS_W;
            int gi = reflect_idx(iy0 + lr, H_IN);
            int gc = reflect_idx(ix0 + lc, W_IN);
            const float* gp = src + gi * W_IN + gc;
            float* lp = &tile[lr * LDS_STRIDE + lc];
#if USE_ASYNC_LDS
            __builtin_amdgcn_global_load_async_to_lds_b32(
                (as1_int*)(uintptr_t)gp,
                (as3_int*)(uint32_t)(uintptr_t)lp,
                0, 0);
#else
            *lp = *gp;
#endif
        }
    }
#if USE_ASYNC_LDS
#if __has_builtin(__builtin_amdgcn_s_wait_asynccnt)
    __builtin_amdgcn_s_wait_asynccnt(0);
#else
    asm volatile("s_wait_asynccnt 0" ::: "memory");
#endif
#endif
    __syncthreads();

    // ---- Compute: each thread -> 4 rows x one float4 ---------------------
    const int tx  = tid & 31;   // float4 column within the 128-wide tile
    const int tyb = tid >> 5;   // base output row (0..7)
    const int c   = tx * 2;     // LDS column of v0 (input col n-1)

    float* __restrict__ dstBase =
        out + (size_t)img * (H_OUT * W_OUT) + ox0 + tx * 4;

    #pragma unroll
    for (int r = 0; r < 4; ++r) {
        const int oyl = tyb + 8 * r;          // local output row 0..31
        const int p   = oyl & 1;              // phase
        const int lr0 = (oyl >> 1) + p;       // first LDS row of the pair
        const float wA = p ? 0.75f : 0.25f;
        const float wB = p ? 0.25f : 0.75f;

        const float* ra = &tile[lr0 * LDS_STRIDE + c];
        const float* rb = ra + LDS_STRIDE;

        const float v0 = wA * ra[0] + wB * rb[0];
        const float v1 = wA * ra[1] + wB * rb[1];
        const float v2 = wA * ra[2] + wB * rb[2];
        const float v3 = wA * ra[3] + wB * rb[3];

        float4 o;
        o.x = 0.25f * v0 + 0.75f * v1;
        o.y = 0.75f * v1 + 0.25f * v2;
        o.z = 0.25f * v1 + 0.75f * v2;
        o.w = 0.75f * v2 + 0.25f * v3;

        *(float4*)(dstBase + (size_t)(oy0 + oyl) * W_OUT) = o;
    }
}

extern "C" void kernel_launch(void* const* d_in, const int* in_sizes, int n_in,
                              void* d_out, int out_size, void* d_ws, size_t ws_size,
                              hipStream_t stream) {
    (void)n_in; (void)d_ws; (void)ws_size; (void)out_size;
    const float* in = (const float*)d_in[0];
    // d_in[1] is the fixed [1,3,3,1] outer-product kernel; its (normalized)
    // separable weights {0.25, 0.75} are baked into the kernel above.
    float* out = (float*)d_out;

    const int images = in_sizes[0] / (H_IN * W_IN);  // N*C = 256
    dim3 grid(W_OUT / TILE_OW, H_OUT / TILE_OH, images);
    dim3 block(256);
    upfirdn2d_up2_kernel<<<grid, block, 0, stream>>>(in, out);
}